// TimeSeriesTransformersSelfAttention_83880711291148
// MI455X (gfx1250) — compile-verified
//
#include <hip/hip_runtime.h>
#include <math.h>

#define BATCH 64
#define HEADS 8
#define HDIM  64
#define EMB   512
#define NBLK  32
#define BLKSZ 128
#define SEQL  4096   // NBLK * BLKSZ

typedef __attribute__((ext_vector_type(2))) float v2f;
typedef __attribute__((ext_vector_type(8))) float v8f;

// ---------------------------------------------------------------------------
// Y[M,N] = (X[M,K] @ W[N,K]^T + bias[N]) * scale, via V_WMMA_F32_16X16X4_F32.
// One wave32 per 16x16 output tile. EXEC is all-ones (blockDim=32, no
// divergence) as WMMA requires.
//   A (16x4 f32): lane m(0..15)/half: VGPR j holds A[m][2*half + j]
//   B (4x16 f32): lane n(0..15)/half: VGPR j holds B[2*half + j][n]
//   C/D (16x16):  VGPR r: lanes 0-15 -> row r, lanes 16-31 -> row r+8
// ---------------------------------------------------------------------------
__global__ __launch_bounds__(32) void gemm16_xwT(const float* __restrict__ X,
                                                 const float* __restrict__ W,
                                                 const float* __restrict__ bias,
                                                 float* __restrict__ Y,
                                                 float scale)
{
    const int K = EMB, N = EMB;
    const int lane = threadIdx.x;
    const int mr   = lane & 15;
    const int half = lane >> 4;

    const float* xp = X + (size_t)(blockIdx.x * 16 + mr) * K + 2 * half;
    const float* wp = W + (size_t)(blockIdx.y * 16 + mr) * K + 2 * half;

    v8f acc = {};
    #pragma unroll 4
    for (int k = 0; k < K; k += 4) {
        v2f a = *(const v2f*)(xp + k);   // 8B aligned: (2*half + k) is even
        v2f b = *(const v2f*)(wp + k);
        acc = __builtin_amdgcn_wmma_f32_16x16x4_f32(
            /*neg_a=*/false, a, /*neg_b=*/false, b,
            /*c_mod=*/(short)0, acc, /*reuse_a=*/false, /*reuse_b=*/false);
    }

    const int col  = blockIdx.y * 16 + mr;
    const float bv = bias[col];
    const int row0 = blockIdx.x * 16 + half * 8;
    #pragma unroll
    for (int r = 0; r < 8; ++r)
        Y[(size_t)(row0 + r) * N + col] = (acc[r] + bv) * scale;
}

// ---------------------------------------------------------------------------
// Paged-attention decode for one (b,h) per 256-thread block.
// Streams K once (score pass), V once (context pass); softmax kept in LDS.
// ---------------------------------------------------------------------------
__global__ __launch_bounds__(256) void paged_attn(
    const float* __restrict__ qbuf,    // [B,H,D] pre-scaled q
    const float* __restrict__ knew,    // [B,H,D]
    const float* __restrict__ vnew,    // [B,H,D]
    const float* __restrict__ kcache,  // [B,H,NBLK,BLKSZ,D]
    const float* __restrict__ vcache,  // [B,H,NBLK,BLKSZ,D]
    const float* __restrict__ mask,    // [B,1,1,SEQL] additive
    const int*  __restrict__ cpos,     // scalar
    const int*  __restrict__ btab,     // [B,NBLK]
    float* __restrict__ ctx)           // [B,H,D]
{
    __shared__ float s_scores[SEQL];   // 16 KB
    __shared__ float s_q[HDIM];
    __shared__ int   s_pb[NBLK];
    __shared__ float s_rmax[8];
    __shared__ float s_rsum[8];
    __shared__ float s_part[256];

    const int b   = blockIdx.x;
    const int h   = blockIdx.y;
    const int tid = threadIdx.x;
    const int pos = cpos[0];
    const int bh  = b * HEADS + h;

    if (tid < HDIM) s_q[tid]  = qbuf[bh * HDIM + tid];
    if (tid < NBLK) s_pb[tid] = btab[b * NBLK + tid];
    __syncthreads();

    const float* kb   = kcache + (size_t)bh * (NBLK * BLKSZ * HDIM);
    const float* vb   = vcache + (size_t)bh * (NBLK * BLKSZ * HDIM);
    const float* mrow = mask + (size_t)b * SEQL;

    // ---- score pass: one KV row (256B contiguous) per thread per step ----
    for (int l = tid; l < SEQL; l += 256) {
        const float* kr;
        if (l == pos) {
            kr = knew + bh * HDIM;                       // paged-cache update
        } else {
            const int pb = s_pb[l >> 7];
            kr = kb + (size_t)(pb * BLKSZ + (l & (BLKSZ - 1))) * HDIM;
        }
        float acc = 0.f;
        const float4* kr4 = (const float4*)kr;
        #pragma unroll
        for (int d4 = 0; d4 < HDIM / 4; ++d4) {
            float4 kv = kr4[d4];
            acc = fmaf(s_q[4 * d4 + 0], kv.x, acc);
            acc = fmaf(s_q[4 * d4 + 1], kv.y, acc);
            acc = fmaf(s_q[4 * d4 + 2], kv.z, acc);
            acc = fmaf(s_q[4 * d4 + 3], kv.w, acc);
        }
        s_scores[l] = acc + mrow[l];
    }
    __syncthreads();

    // ---- softmax max (wave32 shfl + LDS) ----
    float m = -3.4e38f;
    for (int l = tid; l < SEQL; l += 256) m = fmaxf(m, s_scores[l]);
    #pragma unroll
    for (int off = 16; off > 0; off >>= 1) m = fmaxf(m, __shfl_xor(m, off, 32));
    if ((tid & 31) == 0) s_rmax[tid >> 5] = m;
    __syncthreads();
    float mx = s_rmax[0];
    #pragma unroll
    for (int w = 1; w < 8; ++w) mx = fmaxf(mx, s_rmax[w]);

    // ---- exp + sum ----
    float lsum = 0.f;
    for (int l = tid; l < SEQL; l += 256) {
        float e = __expf(s_scores[l] - mx);
        s_scores[l] = e;
        lsum += e;
    }
    #pragma unroll
    for (int off = 16; off > 0; off >>= 1) lsum += __shfl_xor(lsum, off, 32);
    if ((tid & 31) == 0) s_rsum[tid >> 5] = lsum;
    __syncthreads();
    float total = 0.f;
    #pragma unroll
    for (int w = 0; w < 8; ++w) total += s_rsum[w];
    const float inv = 1.0f / total;

    // ---- context pass: lanes d=0..63 read contiguous 256B V rows ----
    const int d    = tid & 63;
    const int part = tid >> 6;                 // 4 L-partitions
    const int l0   = part * (SEQL / 4);
    float acc = 0.f;
    for (int l = l0; l < l0 + SEQL / 4; ++l) {
        const float wt = s_scores[l];
        float vv;
        if (l == pos) {                        // uniform per 64-thread group
            vv = vnew[bh * HDIM + d];
        } else {
            const int pb = s_pb[l >> 7];
            vv = vb[(size_t)(pb * BLKSZ + (l & (BLKSZ - 1))) * HDIM + d];
        }
        acc = fmaf(wt, vv, acc);
    }
    s_part[tid] = acc;                         // tid == part*64 + d
    __syncthreads();
    if (tid < HDIM) {
        float t = s_part[tid] + s_part[tid + 64] + s_part[tid + 128] + s_part[tid + 192];
        ctx[bh * HDIM + tid] = t * inv;
    }
}

// ---------------------------------------------------------------------------
extern "C" void kernel_launch(void* const* d_in, const int* in_sizes, int n_in,
                              void* d_out, int out_size, void* d_ws, size_t ws_size,
                              hipStream_t stream)
{
    const float* hidden = (const float*)d_in[0];
    const float* q_w    = (const float*)d_in[1];
    const float* q_b    = (const float*)d_in[2];
    const float* k_w    = (const float*)d_in[3];
    const float* k_b    = (const float*)d_in[4];
    const float* v_w    = (const float*)d_in[5];
    const float* v_b    = (const float*)d_in[6];
    const float* o_w    = (const float*)d_in[7];
    const float* o_b    = (const float*)d_in[8];
    const float* kcache = (const float*)d_in[9];
    const float* vcache = (const float*)d_in[10];
    const float* mask   = (const float*)d_in[11];
    const int*   cpos   = (const int*)d_in[12];
    const int*   btab   = (const int*)d_in[13];
    float*       out    = (float*)d_out;

    // scratch: q, k_new, v_new, ctx  (each B*E f32 = 128 KB)
    float* qs = (float*)d_ws;
    float* kn = qs + BATCH * EMB;
    float* vn = kn + BATCH * EMB;
    float* cx = vn + BATCH * EMB;

    const dim3 gg(BATCH / 16, EMB / 16);   // (4, 32) tiles, 32 threads each
    const float qscale = 0.125f;           // D^-0.5 = 64^-0.5

    gemm16_xwT<<<gg, 32, 0, stream>>>(hidden, q_w, q_b, qs, qscale);
    gemm16_xwT<<<gg, 32, 0, stream>>>(hidden, k_w, k_b, kn, 1.0f);
    gemm16_xwT<<<gg, 32, 0, stream>>>(hidden, v_w, v_b, vn, 1.0f);

    paged_attn<<<dim3(BATCH, HEADS), 256, 0, stream>>>(
        qs, kn, vn, kcache, vcache, mask, cpos, btab, cx);

    gemm16_xwT<<<gg, 32, 0, stream>>>(cx, o_w, o_b, out, 1.0f);
}